// Interval_Refine_75788992905543
// MI455X (gfx1250) — compile-verified
//
#include <hip/hip_runtime.h>
#include <math.h>

typedef __attribute__((ext_vector_type(2))) float v2f;
typedef __attribute__((ext_vector_type(8))) float v8f;

#define NN     4096
#define DD     256
#define CC     5
#define G3D    768          // 3*D
#define NITV   70
#define MLPH   256
#define QDIM   261          // D + 5
#define LSTRIDE 160         // fixed per-row logit stride (2*max_bins)

__constant__ __device__ int c_dummy; // keep at least one device symbol

static __device__ __forceinline__ float sigmoidf_(float x) {
    return 1.0f / (1.0f + __expf(-x));
}

struct KParams {
    // inputs
    const float* emb;        // [N,D]
    const float* pred;       // [N,C]
    const float* anc;        // [70,2]
    const float* convw;      // [C,1,5]
    const float* Whh[3];     // [3D,D]
    const float* Wih[3];     // [3D,D]
    const float* aWhh[3];    // [3,1]
    const float* aWih[3];    // [3,1]
    const float* abhh[3];    // [3]
    const float* abih[3];    // [3]
    const float* bhh[3];     // [3D]
    const float* bih[3];     // [3D]
    const float* clsW[3];    // [4,257]
    const float* clsb[3];    // [4]
    const float* confW[3];   // [1,257]
    const float* confb[3];   // [1]
    const float* W1[3][3];   // [256,261]
    const float* W2[3][3];   // [2b,256]
    const float* b1[3][3];   // [256]
    const float* b2[3][3];   // [2b]
    const float* wp[3];      // [b]
    const float* timepos;    // [N,1]
    const float* alen;       // [1]
    // workspace
    float* t;                // [N]
    float* abn;              // [N]
    int*   lo;               // [70]
    int*   hi;               // [70]
    float* hloc;             // [70,256]
    float* hab;              // [70]
    float* logits;           // [3 scales][3 layers][30][160]
    float* gx[3];            // [N, 768]
    // output
    float* out;              // [491]
    int ns[3];               // {30,25,15}
    int offs[3];             // {0,30,55}
    int bins[3];             // {80,60,40}
};

static __device__ __forceinline__ int scale_of(int i) {
    return (i < 30) ? 0 : ((i < 55) ? 1 : 2);
}

// ---------------- K1: t = timepos*alen ; abnormal = 1 - softmax(conv(pred))[0]
__global__ __launch_bounds__(256) void prep_kernel(KParams p) {
    int i = blockIdx.x * blockDim.x + threadIdx.x;
    if (i >= NN) return;
    float al = p.alen[0];
    p.t[i] = p.timepos[i] * al;
    float sm[CC];
    #pragma unroll
    for (int c = 0; c < CC; ++c) {
        float a = 0.f;
        #pragma unroll
        for (int k = 0; k < 5; ++k) {
            int idx = i + k - 2;
            if (idx >= 0 && idx < NN)
                a += p.convw[c * 5 + k] * p.pred[idx * CC + c];
        }
        sm[c] = a;
    }
    float m = sm[0];
    #pragma unroll
    for (int c = 1; c < CC; ++c) m = fmaxf(m, sm[c]);
    float s = 0.f, e0 = 0.f;
    #pragma unroll
    for (int c = 0; c < CC; ++c) {
        float e = __expf(sm[c] - m);
        s += e;
        if (c == 0) e0 = e;
    }
    p.abn[i] = 1.0f - e0 / s;
}

// ---------------- K2: per-interval [lo,hi] via binary search on sorted t
__global__ __launch_bounds__(128) void bounds_kernel(KParams p) {
    int i = threadIdx.x;
    if (i >= NITV) return;
    float st = p.anc[2 * i], en = p.anc[2 * i + 1];
    int a = 0, b = NN;
    while (a < b) { int m = (a + b) >> 1; if (p.t[m] < st) a = m + 1; else b = m; }
    int lo = a;                       // first t >= start
    a = 0; b = NN;
    while (a < b) { int m = (a + b) >> 1; if (p.t[m] <= en) a = m + 1; else b = m; }
    int hi = a - 1;                   // last t <= end
    p.lo[i] = lo;
    p.hi[i] = hi;
}

// ---------------- K3: gx[s] = emb @ Wih[s]^T + bih[s]  via V_WMMA_F32_16X16X4_F32
// One wave computes one 16x16 output tile; K=256 stepped by 4.
// f32 A 16x4: lanes 0-15 hold K{0,1}, lanes 16-31 hold K{2,3} (float2/lane).
// f32 B 4x16: symmetric (K rows striped across VGPRs, half-waves offset by 2).
__global__ __launch_bounds__(256) void gemm_gx_kernel(KParams p) {
    int s = blockIdx.z;
    const float* Wih = p.Wih[s];
    const float* bih = p.bih[s];
    float* gx = p.gx[s];

    int lane = threadIdx.x & 31;
    int wave = threadIdx.x >> 5;
    int half = lane >> 4;
    int nr   = lane & 15;

    int n0 = blockIdx.x * 16;                 // column tile (768 dim)
    int m0 = (blockIdx.y * 8 + wave) * 16;    // row tile (4096 dim)

    float bn = bih[n0 + nr];
    v8f c;
    #pragma unroll
    for (int r = 0; r < 8; ++r) c[r] = bn;    // D = A*B + C chains the bias

    const float* arow = p.emb + (size_t)(m0 + nr) * DD;  // A row for this lane's M
    const float* brow = Wih  + (size_t)(n0 + nr) * DD;   // Wih row n = B column n
    int ko = 2 * half;

    #pragma unroll 8
    for (int k0 = 0; k0 < DD; k0 += 4) {
        v2f a = *(const v2f*)(arow + k0 + ko);
        v2f b = *(const v2f*)(brow + k0 + ko);
        c = __builtin_amdgcn_wmma_f32_16x16x4_f32(
                /*neg_a=*/false, a, /*neg_b=*/false, b,
                /*c_mod=*/(short)0, c, /*reuse_a=*/false, /*reuse_b=*/false);
    }
    #pragma unroll
    for (int r = 0; r < 8; ++r) {
        int m = m0 + r + 8 * half;
        gx[(size_t)m * G3D + n0 + nr] = c[r];
    }
}

// ---------------- K4: per-interval GRU recurrence over [lo,hi]
// 1 block per interval, 768 threads (one Whh row each), h[256] in LDS.
__global__ __launch_bounds__(768) void gru_kernel(KParams p) {
    int i = blockIdx.x;
    int s = scale_of(i);
    int j = threadIdx.x;                    // 0..767
    int lo = p.lo[i], hi = p.hi[i];
    const float* bih = p.bih[s];
    const float* bhh = p.bhh[s];

    if (lo > hi) {                          // empty interval: bias-only step from h0=0
        if (j < DD) {
            float r0 = sigmoidf_(bih[j]        + bhh[j]);
            float z0 = sigmoidf_(bih[DD + j]   + bhh[DD + j]);
            p.hloc[i * DD + j] = (1.f - z0) * tanhf(bih[2 * DD + j] + r0 * bhh[2 * DD + j]);
        }
        return;
    }

    __shared__ float h[DD];
    __shared__ float gh[G3D];
    if (j < DD) h[j] = 0.f;
    float bh = bhh[j];
    const float* wrow = p.Whh[s] + (size_t)j * DD;   // L2-resident (768KB/scale)
    const float* gx = p.gx[s];
    __syncthreads();

    for (int st = lo; st <= hi; ++st) {
        const float* g = gx + (size_t)st * G3D;
        __builtin_prefetch(g + G3D, 0, 3);
        float acc = 0.f;
        #pragma unroll 8
        for (int k = 0; k < DD; k += 4) {
            float4 hv = *(const float4*)&h[k];
            float4 wv = *(const float4*)(wrow + k);
            acc += hv.x * wv.x + hv.y * wv.y + hv.z * wv.z + hv.w * wv.w;
        }
        gh[j] = acc + bh;
        __syncthreads();
        if (j < DD) {
            float r    = sigmoidf_(g[j]        + gh[j]);
            float z    = sigmoidf_(g[DD + j]   + gh[DD + j]);
            float cand = tanhf(g[2 * DD + j] + r * gh[2 * DD + j]);
            h[j] = (1.f - z) * cand + z * h[j];   // only thread j touches h[j] here
        }
        __syncthreads();
    }
    if (j < DD) p.hloc[i * DD + j] = h[j];
}

// ---------------- K5: scalar abnormal-GRU (H=1), one thread per interval
__global__ __launch_bounds__(128) void agru_kernel(KParams p) {
    int i = threadIdx.x;
    if (i >= NITV) return;
    int s = scale_of(i);
    const float* wi = p.aWih[s];
    const float* wh = p.aWhh[s];
    const float* bi = p.abih[s];
    const float* bh = p.abhh[s];
    int lo = p.lo[i], hi = p.hi[i];
    float h = 0.f;
    if (lo > hi) {
        float r0 = sigmoidf_(bi[0] + bh[0]);
        float z0 = sigmoidf_(bi[1] + bh[1]);
        h = (1.f - z0) * tanhf(bi[2] + r0 * bh[2]);
    } else {
        for (int st = lo; st <= hi; ++st) {
            float a = p.abn[st];
            float r    = sigmoidf_(wi[0] * a + bi[0] + wh[0] * h + bh[0]);
            float z    = sigmoidf_(wi[1] * a + bi[1] + wh[1] * h + bh[1]);
            float cand = tanhf(wi[2] * a + bi[2] + r * (wh[2] * h + bh[2]));
            h = (1.f - z) * cand + z * h;
        }
    }
    p.hab[i] = h;
}

// softmax(logits[0..b)) . wp
static __device__ __forceinline__ float softdot_(const float* l, int b, const float* wp) {
    float m = l[0];
    for (int k = 1; k < b; ++k) m = fmaxf(m, l[k]);
    float s = 0.f, d = 0.f;
    for (int k = 0; k < b; ++k) {
        float e = __expf(l[k] - m);
        s += e;
        d += e * wp[k];
    }
    return d / s;
}

// ---------------- K6: refinement MLP + conf/cls heads; 1 block per scale
__global__ __launch_bounds__(256) void head_kernel(KParams p) {
    int s   = blockIdx.x;
    int tid = threadIdx.x;
    int n = p.ns[s], off = p.offs[s], b = p.bins[s];
    float al = p.alen[0];

    __shared__ float qrow[QDIM + 3];
    __shared__ float hrow[MLPH];
    __shared__ float sst[30], sen[30], sc[30], sw[30];
    __shared__ float lbuf[30 * LSTRIDE];

    if (tid < n) {
        float st = p.anc[2 * (off + tid)], en = p.anc[2 * (off + tid) + 1];
        sst[tid] = st; sen[tid] = en;
        sc[tid] = 0.5f * (st + en);
        sw[tid] = en - st;
    }
    __syncthreads();

    for (int l = 0; l < 3; ++l) {
        const float* W1 = p.W1[s][l];
        const float* B1 = p.b1[s][l];
        const float* W2 = p.W2[s][l];
        const float* B2 = p.b2[s][l];
        for (int i = 0; i < n; ++i) {
            qrow[tid] = p.hloc[(size_t)(off + i) * DD + tid];     // tid<256 == DD
            if (tid == 0) {
                qrow[256] = sc[i] / al;  qrow[257] = sw[i] / al;
                qrow[258] = sst[i] / al; qrow[259] = sen[i] / al;
                qrow[260] = p.hab[off + i];
            }
            __syncthreads();
            {   // hmid row: each thread one hidden unit
                const float* w = W1 + (size_t)tid * QDIM;
                float a = B1[tid];
                #pragma unroll 4
                for (int k = 0; k < QDIM; ++k) a += qrow[k] * w[k];
                hrow[tid] = fmaxf(a, 0.f);
            }
            __syncthreads();
            if (tid < 2 * b) {          // logits row (2b <= 160 <= 256)
                const float* w = W2 + (size_t)tid * MLPH;
                float a = B2[tid];
                #pragma unroll 4
                for (int k = 0; k < MLPH; ++k) a += hrow[k] * w[k];
                lbuf[i * LSTRIDE + tid] = a;
                p.logits[((size_t)s * 3 + l) * 30 * LSTRIDE + i * LSTRIDE + tid] = a;
            }
            __syncthreads();
        }
        if (tid < n) {                  // starts/ends update
            sst[tid] += softdot_(&lbuf[tid * LSTRIDE],     b, p.wp[s]);
            sen[tid] += softdot_(&lbuf[tid * LSTRIDE + b], b, p.wp[s]);
        }
        __syncthreads();
    }

    if (tid < n) {
        int gi = off + tid;
        p.out[2 * gi]     = sst[tid];
        p.out[2 * gi + 1] = sen[tid];
        const float* hl = p.hloc + (size_t)gi * DD;
        float ha = p.hab[gi];
        float cf = p.confb[s][0];
        for (int k = 0; k < DD; ++k) cf += hl[k] * p.confW[s][k];
        cf += ha * p.confW[s][DD];
        p.out[141 + gi] = cf;
        #pragma unroll
        for (int o = 0; o < CC - 1; ++o) {
            const float* w = p.clsW[s] + o * (DD + 1);
            float cl = p.clsb[s][o];
            for (int k = 0; k < DD; ++k) cl += hl[k] * w[k];
            cl += ha * w[DD];
            p.out[211 + gi * (CC - 1) + o] = cl;
        }
    }
}

// ---------------- K7: distillation KL reduction (280 row-KLs total)
__global__ __launch_bounds__(256) void distill_kernel(KParams p) {
    __shared__ float red[256];
    float acc = 0.f;
    for (int item = threadIdx.x; item < 280; item += 256) {
        int i2 = item, s = 0;
        while (i2 >= p.ns[s] * 4) { i2 -= p.ns[s] * 4; ++s; }
        int n = p.ns[s], b = p.bins[s];
        int l = i2 / (2 * n);            // student layer 0 or 1
        int r2 = i2 % (2 * n);
        int which = r2 / n;              // 0: start-logits, 1: end-logits
        int row = r2 % n;
        const float* sl = p.logits + ((size_t)s * 3 + l) * 30 * LSTRIDE + row * LSTRIDE + which * b;
        const float* tl = p.logits + ((size_t)s * 3 + 2) * 30 * LSTRIDE + row * LSTRIDE + which * b;
        float mt = tl[0], ms = sl[0];
        for (int k = 1; k < b; ++k) { mt = fmaxf(mt, tl[k]); ms = fmaxf(ms, sl[k]); }
        float zt = 0.f, zs = 0.f;
        for (int k = 0; k < b; ++k) { zt += __expf(tl[k] - mt); zs += __expf(sl[k] - ms); }
        float lzt = __logf(zt), lzs = __logf(zs);
        float kl = 0.f;
        for (int k = 0; k < b; ++k) {
            float lpt = tl[k] - mt - lzt;
            float lps = sl[k] - ms - lzs;
            kl += __expf(lpt) * (lpt - lps);
        }
        acc += kl / (float)n;
    }
    red[threadIdx.x] = acc;
    __syncthreads();
    for (int w = 128; w > 0; w >>= 1) {
        if (threadIdx.x < w) red[threadIdx.x] += red[threadIdx.x + w];
        __syncthreads();
    }
    if (threadIdx.x == 0) p.out[140] = red[0];
}

extern "C" void kernel_launch(void* const* d_in, const int* in_sizes, int n_in,
                              void* d_out, int out_size, void* d_ws, size_t ws_size,
                              hipStream_t stream) {
    (void)in_sizes; (void)n_in; (void)out_size; (void)ws_size;
    KParams p{};
    // params pytree flattened with dict keys sorted (jax tree order):
    // conv_w, then per scale: Whh,Wih,aWhh,aWih,abhh,abih,bhh,bih,clsW,clsb,
    // confW,confb, layers[l]{W1,W2,b1,b2}, wp
    p.emb   = (const float*)d_in[0];
    p.pred  = (const float*)d_in[1];
    p.anc   = (const float*)d_in[2];
    p.convw = (const float*)d_in[3];
    for (int s = 0; s < 3; ++s) {
        int sb = 4 + 25 * s;
        p.Whh[s]   = (const float*)d_in[sb + 0];
        p.Wih[s]   = (const float*)d_in[sb + 1];
        p.aWhh[s]  = (const float*)d_in[sb + 2];
        p.aWih[s]  = (const float*)d_in[sb + 3];
        p.abhh[s]  = (const float*)d_in[sb + 4];
        p.abih[s]  = (const float*)d_in[sb + 5];
        p.bhh[s]   = (const float*)d_in[sb + 6];
        p.bih[s]   = (const float*)d_in[sb + 7];
        p.clsW[s]  = (const float*)d_in[sb + 8];
        p.clsb[s]  = (const float*)d_in[sb + 9];
        p.confW[s] = (const float*)d_in[sb + 10];
        p.confb[s] = (const float*)d_in[sb + 11];
        for (int l = 0; l < 3; ++l) {
            p.W1[s][l] = (const float*)d_in[sb + 12 + 4 * l + 0];
            p.W2[s][l] = (const float*)d_in[sb + 12 + 4 * l + 1];
            p.b1[s][l] = (const float*)d_in[sb + 12 + 4 * l + 2];
            p.b2[s][l] = (const float*)d_in[sb + 12 + 4 * l + 3];
        }
        p.wp[s] = (const float*)d_in[sb + 24];
    }
    p.timepos = (const float*)d_in[79];
    p.alen    = (const float*)d_in[80];

    // workspace carve (256B aligned chunks)
    char* w = (char*)d_ws;
    auto carve = [&](size_t bytes) -> void* {
        void* r = (void*)w;
        w += (bytes + 255) & ~(size_t)255;
        return r;
    };
    p.t      = (float*)carve((size_t)NN * 4);
    p.abn    = (float*)carve((size_t)NN * 4);
    p.lo     = (int*)  carve((size_t)NITV * 4);
    p.hi     = (int*)  carve((size_t)NITV * 4);
    p.hloc   = (float*)carve((size_t)NITV * DD * 4);
    p.hab    = (float*)carve((size_t)NITV * 4);
    p.logits = (float*)carve((size_t)3 * 3 * 30 * LSTRIDE * 4);
    for (int s = 0; s < 3; ++s)
        p.gx[s] = (float*)carve((size_t)NN * G3D * 4);
    p.out = (float*)d_out;
    p.ns[0] = 30; p.ns[1] = 25; p.ns[2] = 15;
    p.offs[0] = 0; p.offs[1] = 30; p.offs[2] = 55;
    p.bins[0] = 80; p.bins[1] = 60; p.bins[2] = 40;

    prep_kernel   <<<dim3(NN / 256), dim3(256), 0, stream>>>(p);
    bounds_kernel <<<dim3(1),        dim3(128), 0, stream>>>(p);
    gemm_gx_kernel<<<dim3(G3D / 16, NN / 128, 3), dim3(256), 0, stream>>>(p);
    gru_kernel    <<<dim3(NITV),     dim3(768), 0, stream>>>(p);
    agru_kernel   <<<dim3(1),        dim3(128), 0, stream>>>(p);
    head_kernel   <<<dim3(3),        dim3(256), 0, stream>>>(p);
    distill_kernel<<<dim3(1),        dim3(256), 0, stream>>>(p);
}